// SpatialTransformer_70643622084910
// MI455X (gfx1250) — compile-verified
//
#include <hip/hip_runtime.h>
#include <hip/hip_bf16.h>

#define BATCH 256
#define CH    3
#define HIN   256
#define WIN   256
#define HOUT  64
#define WOUT  64
#define ROWS_PER_BLOCK 4   // output rows per block; 64 threads per row group

// ---- gfx1250 async-to-LDS feature detection (probe-confirmed present) -----
#if defined(__has_builtin)
#if __has_builtin(__builtin_amdgcn_global_load_async_to_lds_b128)
#define HAVE_ASYNC_LDS_B128 1
#endif
#if __has_builtin(__builtin_amdgcn_s_wait_asynccnt)
#define HAVE_WAIT_ASYNC 1
#endif
#endif

// Builtin signature (from round-1 diagnostic):
//   (v4i addrspace(1)* src, v4i addrspace(3)* dst, int offset, int cpol)
typedef int v4i __attribute__((__vector_size__(16)));
typedef __attribute__((address_space(1))) v4i* g_v4i_p;
typedef __attribute__((address_space(3))) v4i* l_v4i_p;

__global__ __launch_bounds__(256) void
st_bilinear_kernel(const float* __restrict__ x,
                   const float* __restrict__ zw,
                   const int*   __restrict__ invp,
                   float*       __restrict__ out)
{
    // LDS staging: [row group][y0/y1][channel][WIN] = 4*2*3*256*4B = 24 KB
    __shared__ float lds[ROWS_PER_BLOCK][2][CH][WIN];

    const int b     = blockIdx.y;
    const int iBase = blockIdx.x * ROWS_PER_BLOCK;
    const int tid   = threadIdx.x;
    const int r     = tid >> 6;   // row group 0..3 (2 waves each)
    const int j     = tid & 63;   // output column 0..63
    const int i     = iBase + r;  // output row

    // Per-batch affine params (uniform -> scalar loads)
    float sx = zw[b * 4 + 0];
    float sy = zw[b * 4 + 1];
    float tx = zw[b * 4 + 2];
    float ty = zw[b * 4 + 3];
    if (invp[0]) {
        float s0 = sx + 1e-6f;
        float s1 = sy + 1e-6f;
        sx = 1.0f / s0;
        sy = 1.0f / s1;
        tx = -tx / s0;
        ty = -ty / s1;
    }

    // Vertical source coordinate for this output row
    float ysv = (2.0f * (float)i + 1.0f) / (float)HOUT - 1.0f;
    float v   = sy * ysv + ty;
    float iy  = ((v + 1.0f) * (float)HIN - 1.0f) * 0.5f;
    float fy0 = floorf(iy);
    int   y0  = (int)fy0;
    int   y1  = y0 + 1;
    float wy1 = iy - fy0;
    float wy0 = 1.0f - wy1;

    // ---- Issue DMA: stage rows y0,y1 (all channels) into LDS --------------
    // 64 threads per row group; each moves one b128 per (yy, ch).
    bool issued_async = false;
    #pragma unroll
    for (int yy = 0; yy < 2; ++yy) {
        int  ysrc  = (yy == 0) ? y0 : y1;
        bool valid = (ysrc >= 0) && (ysrc < HIN);   // uniform per row group
        #pragma unroll
        for (int c = 0; c < CH; ++c) {
            float* dst = &lds[r][yy][c][j * 4];
            if (valid) {
                const float* src =
                    x + (((size_t)(b * CH + c) * HIN + (size_t)ysrc) * WIN + (size_t)j * 4);
#if defined(HAVE_ASYNC_LDS_B128)
                // Async DMA global->LDS, tracked by ASYNCcnt (gfx1250 path).
                __builtin_amdgcn_global_load_async_to_lds_b128(
                    (g_v4i_p)(v4i*)(uintptr_t)src,   // strip const, then AS1
                    (l_v4i_p)(v4i*)dst,              // generic -> AS3
                    0, 0);
                issued_async = true;
#else
                *(float4*)dst = *(const float4*)src;
#endif
            } else {
                float4 z4 = make_float4(0.f, 0.f, 0.f, 0.f);
                *(float4*)dst = z4;   // zeros padding in y
            }
        }
    }

    // ---- Overlap: horizontal coordinate math while the DMA is in flight ---
    float xsv = (2.0f * (float)j + 1.0f) / (float)WOUT - 1.0f;
    float u   = sx * xsv + tx;
    float ix  = ((u + 1.0f) * (float)WIN - 1.0f) * 0.5f;
    float fx0 = floorf(ix);
    int   x0  = (int)fx0;
    int   x1  = x0 + 1;
    float wx1 = ix - fx0;
    float wx0 = 1.0f - wx1;

    bool vx0 = (x0 >= 0) && (x0 < WIN);
    bool vx1 = (x1 >= 0) && (x1 < WIN);
    int  x0c = min(max(x0, 0), WIN - 1);
    int  x1c = min(max(x1, 0), WIN - 1);
    float m0 = vx0 ? wx0 : 0.0f;   // zeros padding in x folded into weights
    float m1 = vx1 ? wx1 : 0.0f;

    const size_t cstride = (size_t)HOUT * WOUT;
    size_t obase = (size_t)b * CH * cstride + (size_t)i * WOUT + (size_t)j;

    // ---- Drain DMA, then block barrier -----------------------------------
#if defined(HAVE_ASYNC_LDS_B128)
    if (issued_async) {
#if defined(HAVE_WAIT_ASYNC)
        __builtin_amdgcn_s_wait_asynccnt(0);
#else
        asm volatile("s_wait_asynccnt 0x0" ::: "memory");
#endif
    }
#else
    (void)issued_async;
#endif
    __syncthreads();

    // ---- Bilinear blend from LDS, non-temporal output stores -------------
    #pragma unroll
    for (int c = 0; c < CH; ++c) {
        float a00 = lds[r][0][c][x0c];
        float a01 = lds[r][0][c][x1c];
        float a10 = lds[r][1][c][x0c];
        float a11 = lds[r][1][c][x1c];
        // invalid y rows were zero-filled in LDS -> zeros padding in y
        float res = wy0 * (m0 * a00 + m1 * a01) + wy1 * (m0 * a10 + m1 * a11);
        __builtin_nontemporal_store(res, out + obase + (size_t)c * cstride);
    }
}

extern "C" void kernel_launch(void* const* d_in, const int* in_sizes, int n_in,
                              void* d_out, int out_size, void* d_ws, size_t ws_size,
                              hipStream_t stream) {
    const float* x   = (const float*)d_in[0];   // (256, 3, 256, 256) fp32
    const float* zw  = (const float*)d_in[1];   // (256, 4) fp32
    const int*   inv = (const int*)d_in[2];     // scalar int flag
    float*       out = (float*)d_out;           // (256, 3, 64, 64) fp32

    dim3 grid(HOUT / ROWS_PER_BLOCK, BATCH);    // (16, 256)
    dim3 block(256);
    st_bilinear_kernel<<<grid, block, 0, stream>>>(x, zw, inv, out);
}